// Model_77421080477846
// MI455X (gfx1250) — compile-verified
//
#include <hip/hip_runtime.h>
#include <cstdint>

#define NDIM        4096
#define BATCH       4
#define ROWS_PER_WG 32
#define TILES_PER_B (NDIM / ROWS_PER_WG)   // 128
#define SINK_ITERS  20
#define SINK_EPS    1e-6f
#define TPB         256

typedef __attribute__((ext_vector_type(2))) float v2f;
typedef __attribute__((ext_vector_type(8))) float v8f;

// ---- CDNA5 async global->LDS helpers (gfx1250 async-tensor path) ----------
__device__ __forceinline__ uint32_t lds_off(const void* p) {
  // generic (flat) LDS address: low 32 bits == wave-relative LDS byte offset
  return (uint32_t)(uintptr_t)p;
}

__device__ __forceinline__ void async_load_b128(uint32_t lds_byte,
                                                uint32_t glb_byte_off,
                                                const void* base) {
  // GVS mode: global address is the 64-bit SGPR base plus the 32-bit VGPR offset; LDS address from the first VGPR
  asm volatile("global_load_async_to_lds_b128 %0, %1, %2"
               :: "v"(lds_byte), "v"(glb_byte_off), "s"(base)
               : "memory");
}
__device__ __forceinline__ void wait_async_0() {
  asm volatile("s_wait_asynccnt 0" ::: "memory");
}
__device__ __forceinline__ void wait_async_4() {
  asm volatile("s_wait_asynccnt 4" ::: "memory");
}

// ---- prep: rowmax + E = exp(x - rowmax) written into d_out; r = 1 ---------
__global__ __launch_bounds__(TPB) void k_prep(const float* __restrict__ x,
                                              float* __restrict__ E,
                                              float* __restrict__ r) {
  __shared__ float red[8];
  const int t   = threadIdx.x;
  const int row = blockIdx.x;                       // b*4096 + i, 0..16383
  const float4* xr = (const float4*)x + (size_t)row * (NDIM / 4);
  float4*       er = (float4*)E       + (size_t)row * (NDIM / 4);

  float4 v[4];
  float m = -3.402823466e+38f;
#pragma unroll
  for (int q = 0; q < 4; ++q) {
    v[q] = xr[t + TPB * q];
    m = fmaxf(m, fmaxf(fmaxf(v[q].x, v[q].y), fmaxf(v[q].z, v[q].w)));
  }
#pragma unroll
  for (int off = 16; off > 0; off >>= 1) m = fmaxf(m, __shfl_xor(m, off, 32));
  if ((t & 31) == 0) red[t >> 5] = m;
  __syncthreads();
  float M = red[0];
#pragma unroll
  for (int w = 1; w < 8; ++w) M = fmaxf(M, red[w]);

#pragma unroll
  for (int q = 0; q < 4; ++q) {
    float4 e;
    e.x = __expf(v[q].x - M);
    e.y = __expf(v[q].y - M);
    e.z = __expf(v[q].z - M);
    e.w = __expf(v[q].w - M);
    er[t + TPB * q] = e;
  }
  if (t == 0) r[row] = 1.0f;
}

__global__ __launch_bounds__(TPB) void k_initc(float* __restrict__ c) {
  const int g = blockIdx.x * TPB + threadIdx.x;
  if (g < BATCH * NDIM) c[g] = 1.0f;
}

// ---- fused Sinkhorn step: row-normalize update of r + column partials -----
// One WG owns 32 full rows of one batch. Rows streamed into LDS with
// double-buffered async b128 copies; c vector async-loaded to LDS once.
__global__ __launch_bounds__(TPB) void k_step(const float* __restrict__ E,
                                              float* __restrict__ r,
                                              const float* __restrict__ c,
                                              float* __restrict__ colpart) {
  __shared__ float c_lds[NDIM];            // 16 KB
  __shared__ float rowbuf[2][NDIM];        // 32 KB (double buffer)
  __shared__ float red[8];

  const int t    = threadIdx.x;
  const int b    = blockIdx.x / TILES_PER_B;
  const int tile = blockIdx.x % TILES_PER_B;
  const int i0   = tile * ROWS_PER_WG;
  const float* cb = c + b * NDIM;
  const float* Eb = E + ((size_t)(b * NDIM + i0)) * NDIM;

  uint32_t off_b[4];
#pragma unroll
  for (int k = 0; k < 4; ++k) off_b[k] = (uint32_t)t * 16u + (uint32_t)k * 4096u;
  const uint32_t lds_c  = lds_off(&c_lds[0]);
  const uint32_t lds_r0 = lds_off(&rowbuf[0][0]);
  const uint32_t lds_r1 = lds_off(&rowbuf[1][0]);

  // async-load c vector (16 KB) and row 0
#pragma unroll
  for (int k = 0; k < 4; ++k) async_load_b128(lds_c + off_b[k], off_b[k], cb);
#pragma unroll
  for (int k = 0; k < 4; ++k) async_load_b128(lds_r0 + off_b[k], off_b[k], Eb);
  wait_async_0();
  __syncthreads();

  float4 cv[4], cp[4];
#pragma unroll
  for (int q = 0; q < 4; ++q) {
    cv[q] = *(const float4*)&c_lds[4 * t + 1024 * q];
    cp[q] = make_float4(0.f, 0.f, 0.f, 0.f);
  }

  for (int i = 0; i < ROWS_PER_WG; ++i) {
    // prefetch row i+1 into the other buffer while we consume row i
    if (i + 1 < ROWS_PER_WG) {
      const uint32_t nxt = ((i + 1) & 1) ? lds_r1 : lds_r0;
      const float* Erow = Eb + (size_t)(i + 1) * NDIM;
#pragma unroll
      for (int k = 0; k < 4; ++k) async_load_b128(nxt + off_b[k], off_b[k], Erow);
      wait_async_4();                       // row i's 4 copies (older) are done
    } else {
      wait_async_0();
    }
    __syncthreads();                        // barrier A: row i visible to all waves

    const float* rb = (i & 1) ? rowbuf[1] : rowbuf[0];
    float4 ev[4];
    float p = 0.f;
#pragma unroll
    for (int q = 0; q < 4; ++q) {
      ev[q] = *(const float4*)&rb[4 * t + 1024 * q];
      p = fmaf(ev[q].x, cv[q].x, p);
      p = fmaf(ev[q].y, cv[q].y, p);
      p = fmaf(ev[q].z, cv[q].z, p);
      p = fmaf(ev[q].w, cv[q].w, p);
    }
    // read r_old BEFORE barrier B; the single writer stores AFTER barrier B
    const float rold = r[b * NDIM + i0 + i];

#pragma unroll
    for (int off = 16; off > 0; off >>= 1) p += __shfl_xor(p, off, 32);
    if ((t & 31) == 0) red[t >> 5] = p;
    __syncthreads();                        // barrier B

    float T = red[0];
#pragma unroll
    for (int w = 1; w < 8; ++w) T += red[w];
    const float rnew = rold / fmaf(rold, T, SINK_EPS);
    if (t == 0) r[b * NDIM + i0 + i] = rnew;

#pragma unroll
    for (int q = 0; q < 4; ++q) {           // column partials (ev still in regs)
      cp[q].x = fmaf(ev[q].x, rnew, cp[q].x);
      cp[q].y = fmaf(ev[q].y, rnew, cp[q].y);
      cp[q].z = fmaf(ev[q].z, rnew, cp[q].z);
      cp[q].w = fmaf(ev[q].w, rnew, cp[q].w);
    }
  }

  // deterministic column reduction: write per-tile partials (summed in fixed
  // order by k_cupdate) instead of FP atomics
  float4* cpo = (float4*)(colpart + ((size_t)(b * TILES_PER_B + tile)) * NDIM);
#pragma unroll
  for (int q = 0; q < 4; ++q) cpo[t + TPB * q] = cp[q];
}

// ---- c update: U = ones(1x128) x colpart(128x4096) done with WMMA --------
// One wave per 16-column block. A == all-ones 16x4 makes D[m][n] = sum_k B[k][n]
// regardless of A striping; 32 chained V_WMMA_F32_16X16X4_F32 accumulate the
// 128 tile partials. C/D layout: lane -> column N, rows M all identical.
__global__ __launch_bounds__(TPB) void k_cupdate(float* __restrict__ c,
                                                 const float* __restrict__ colpart) {
  const int t    = threadIdx.x;
  const int lane = t & 31;
  const int wave = blockIdx.x * (TPB / 32) + (t >> 5);  // 0..1023
  const int b    = wave >> 8;                           // 256 blocks per batch
  const int j0   = (wave & 255) * 16;
  const int n    = lane & 15;
  const int kr   = (lane >> 4) * 2;   // lanes 0-15: K=0,1 ; lanes 16-31: K=2,3

  const float* base = colpart + (size_t)b * TILES_PER_B * NDIM + j0 + n;

  v2f a; a[0] = 1.0f; a[1] = 1.0f;    // all-ones A matrix (16x4)
  v8f d = {};
  for (int k0 = 0; k0 < TILES_PER_B; k0 += 4) {
    v2f bf;
    bf[0] = base[(size_t)(k0 + kr)     * NDIM];
    bf[1] = base[(size_t)(k0 + kr + 1) * NDIM];
    d = __builtin_amdgcn_wmma_f32_16x16x4_f32(false, a, false, bf,
                                              (short)0, d, false, false);
  }
  const float s = d[0];               // U[j0+n], replicated across rows
  if (lane < 16) {
    const int g = b * NDIM + j0 + n;
    const float cj = c[g];
    c[g] = cj / fmaf(cj, s, SINK_EPS);
  }
}

// ---- final: out[i,j] = E[i,j] * r[i] * c[j]  (in place on d_out) ----------
__global__ __launch_bounds__(TPB) void k_final(float* __restrict__ E,
                                               const float* __restrict__ r,
                                               const float* __restrict__ c) {
  const size_t q = (size_t)blockIdx.x * TPB + threadIdx.x;  // float4 index
  const int row = (int)(q >> 10);                           // b*4096 + i
  const int b   = row >> 12;
  const int j4  = (int)(q & 1023);
  float4 v = ((float4*)E)[q];
  const float  ri = r[row];
  const float4 cj = ((const float4*)c)[b * (NDIM / 4) + j4];
  v.x *= ri * cj.x; v.y *= ri * cj.y; v.z *= ri * cj.z; v.w *= ri * cj.w;
  ((float4*)E)[q] = v;
}

extern "C" void kernel_launch(void* const* d_in, const int* in_sizes, int n_in,
                              void* d_out, int out_size, void* d_ws, size_t ws_size,
                              hipStream_t stream) {
  (void)in_sizes; (void)n_in; (void)out_size; (void)ws_size;
  const float* x = (const float*)d_in[0];
  float* E = (float*)d_out;                    // E lives in d_out, scaled in place
  float* ws = (float*)d_ws;                    // needs 8.125 MiB
  float* r       = ws;                         // 16384 floats
  float* c       = ws + BATCH * NDIM;          // 16384 floats
  float* colpart = ws + 2 * BATCH * NDIM;      // 4*128*4096 floats = 8 MiB

  k_prep <<<BATCH * NDIM, TPB, 0, stream>>>(x, E, r);
  k_initc<<<(BATCH * NDIM) / TPB, TPB, 0, stream>>>(c);

  for (int it = 0; it < SINK_ITERS; ++it) {
    k_step   <<<BATCH * TILES_PER_B, TPB, 0, stream>>>(E, r, c, colpart);
    k_cupdate<<<(BATCH * NDIM) / (TPB / 32) / 16, TPB, 0, stream>>>(c, colpart);
  }

  const int nq = (BATCH * NDIM * NDIM) / 4;    // 16,777,216 float4s
  k_final<<<nq / TPB, TPB, 0, stream>>>(E, r, c);
}